// DigitDETR_84997402788107
// MI455X (gfx1250) — compile-verified
//
#include <hip/hip_runtime.h>
#include <hip/hip_bf16.h>
#include <math.h>

// ---------------------------------------------------------------------------
// MSDA (Deformable-DETR) for MI455X / gfx1250.
// GEMMs use V_WMMA_F32_16X16X4_F32 (fp32 in/out -> exact vs fp32 reference).
// ---------------------------------------------------------------------------

typedef __attribute__((ext_vector_type(2))) float v2f;
typedef __attribute__((ext_vector_type(8))) float v8f;

#define D_MODEL 256
#define NHEAD   8
#define NLVL    4
#define NPT     4
#define HEADD   32
#define NQ_C    11531
#define NV_C    11531
#define B_C     2

__constant__ const int   SP_H[4]   = {76, 38, 19, 10};
__constant__ const int   SP_W[4]   = {114, 57, 29, 15};
__constant__ const int   STARTS[4] = {0, 8664, 10830, 11381};

// ---------------------------------------------------------------------------
// WMMA f32 GEMM: C[M,N] = A[M,K] @ B[K,N] + bias[N]   (row-major everywhere)
// One wave computes a 64x32 tile: 4x2 grid of 16x16 WMMA accumulators.
// ---------------------------------------------------------------------------
__global__ __launch_bounds__(256) void gemm_wmma_f32(
    const float* __restrict__ A, const float* __restrict__ Bw,
    const float* __restrict__ bias, float* __restrict__ C,
    int M, int N, int K, int tilesM, int tilesN)
{
    const int wid  = (blockIdx.x * blockDim.x + threadIdx.x) >> 5;
    if (wid >= tilesM * tilesN) return;
    const int tm = wid / tilesN;
    const int tn = wid % tilesN;

    const int lane  = threadIdx.x & 31;
    const int half  = lane >> 4;   // 0: lanes 0-15, 1: lanes 16-31
    const int row16 = lane & 15;

    v8f acc[4][2];
#pragma unroll
    for (int i = 0; i < 4; ++i)
#pragma unroll
        for (int j = 0; j < 2; ++j)
            acc[i][j] = (v8f){0.f, 0.f, 0.f, 0.f, 0.f, 0.f, 0.f, 0.f};

#pragma unroll 2
    for (int k0 = 0; k0 < K; k0 += 4) {
        const int kk = k0 + 2 * half;           // this lane's K pair

        // A fragments: A[m][kk], A[m][kk+1]
        v2f afr[4];
#pragma unroll
        for (int i = 0; i < 4; ++i) {
            const int m  = tm * 64 + i * 16 + row16;
            const bool ok = (m < M);
            const int mc = ok ? m : (M - 1);
            const float* ap = A + (size_t)mc * K + kk;
            float x = ap[0];
            float y = ap[1];
            afr[i].x = ok ? x : 0.f;
            afr[i].y = ok ? y : 0.f;
        }

        // B fragments: B[kk][n], B[kk+1][n]
        v2f bfr[2];
#pragma unroll
        for (int j = 0; j < 2; ++j) {
            const int n = tn * 32 + j * 16 + row16;
            bfr[j].x = Bw[(size_t)kk * N + n];
            bfr[j].y = Bw[(size_t)(kk + 1) * N + n];
        }

#pragma unroll
        for (int i = 0; i < 4; ++i)
#pragma unroll
            for (int j = 0; j < 2; ++j)
                acc[i][j] = __builtin_amdgcn_wmma_f32_16x16x4_f32(
                    false, afr[i], false, bfr[j], (short)0, acc[i][j],
                    false, false);
    }

    // Store: VGPR r of a 16x16 tile -> M = r + 8*half, N = row16
#pragma unroll
    for (int j = 0; j < 2; ++j) {
        const int n = tn * 32 + j * 16 + row16;
        const float bv = bias[n];
#pragma unroll
        for (int i = 0; i < 4; ++i) {
#pragma unroll
            for (int r = 0; r < 8; ++r) {
                const int m = tm * 64 + i * 16 + r + 8 * half;
                if (m < M) C[(size_t)m * N + n] = acc[i][j][r] + bv;
            }
        }
    }
}

// ---------------------------------------------------------------------------
// Sampling kernel: one wave per (b, q, h); lane = channel d (HEADD == 32).
// vbuf   : [B*NV, 256]  projected values (head h at cols h*32..h*32+31)
// offbuf : [B*NQ, 256]  offsets  (head h occupies 32 consecutive floats)
// attnbuf: [B*NQ, 128]  attention logits (head h at cols h*16..h*16+15)
// refp   : [B*NQ, L, 2]
// outbuf : [B*NQ, 256]  head-concatenated result
// ---------------------------------------------------------------------------
__global__ __launch_bounds__(256) void msda_sample_kernel(
    const float* __restrict__ vbuf, const float* __restrict__ offbuf,
    const float* __restrict__ attnbuf, const float* __restrict__ refp,
    float* __restrict__ outbuf)
{
    const int wid = (blockIdx.x * blockDim.x + threadIdx.x) >> 5;
    const int total = B_C * NQ_C * NHEAD;
    if (wid >= total) return;

    const int h  = wid % NHEAD;
    const int bq = wid / NHEAD;       // b*NQ + q
    const int b  = bq / NQ_C;

    const int lane = threadIdx.x & 31;

    // ---- softmax over the 16 (l,p) logits of this head -------------------
    float logit = attnbuf[(size_t)bq * (NHEAD * NLVL * NPT) + h * 16 + (lane & 15)];
    float mx = logit;
#pragma unroll
    for (int s = 8; s >= 1; s >>= 1) mx = fmaxf(mx, __shfl_xor(mx, s, 32));
    float ev = __expf(logit - mx);
    float sum = ev;
#pragma unroll
    for (int s = 8; s >= 1; s >>= 1) sum += __shfl_xor(sum, s, 32);
    const float attn_val = ev / sum;   // lane i (i<16) holds softmax[(l*P+p)=i]

    // ---- per-head offsets: 32 consecutive floats -------------------------
    const float offv = offbuf[(size_t)bq * D_MODEL + h * 32 + lane];

    // ---- reference points: 8 floats per (b,q) ----------------------------
    const float refv = (lane < 8) ? refp[(size_t)bq * (NLVL * 2) + lane] : 0.f;

    const float* __restrict__ vb =
        vbuf + (size_t)b * NV_C * D_MODEL + h * HEADD + lane;

    float acc = 0.f;
#pragma unroll
    for (int l = 0; l < NLVL; ++l) {
        const float rx = __shfl(refv, 2 * l, 32);
        const float ry = __shfl(refv, 2 * l + 1, 32);
        const float lw = (float)SP_W[l];
        const float lh = (float)SP_H[l];
        const float* __restrict__ fb = vb + (size_t)STARTS[l] * D_MODEL;

#pragma unroll
        for (int p = 0; p < NPT; ++p) {
            const int sp = l * NPT + p;
            const float ox = __shfl(offv, 2 * sp, 32);
            const float oy = __shfl(offv, 2 * sp + 1, 32);

            const float px = (rx + ox / lw) * lw - 0.5f;
            const float py = (ry + oy / lh) * lh - 0.5f;
            const float x0 = floorf(px);
            const float y0 = floorf(py);

            float samp = 0.f;
#pragma unroll
            for (int dx = 0; dx < 2; ++dx) {
#pragma unroll
                for (int dy = 0; dy < 2; ++dy) {
                    const float xi = x0 + (float)dx;
                    const float yi = y0 + (float)dy;
                    const float w =
                        (1.f - fabsf(px - xi)) * (1.f - fabsf(py - yi));
                    const bool valid = (xi >= 0.f) && (xi <= lw - 1.f) &&
                                       (yi >= 0.f) && (yi <= lh - 1.f);
                    const float xc = fminf(fmaxf(xi, 0.f), lw - 1.f);
                    const float yc = fminf(fmaxf(yi, 0.f), lh - 1.f);
                    const int idx = (int)(yc * lw + xc);
                    const float g = fb[(size_t)idx * D_MODEL];
                    samp += g * (valid ? w : 0.f);
                }
            }
            acc += samp * __shfl(attn_val, sp, 32);
        }
    }
    outbuf[(size_t)bq * D_MODEL + h * HEADD + lane] = acc;
}

// ---------------------------------------------------------------------------
static inline void launch_gemm(const float* A, const float* Bw, const float* bias,
                               float* C, int M, int N, int K, hipStream_t s)
{
    const int tilesM = (M + 63) / 64;
    const int tilesN = N / 32;
    const int waves  = tilesM * tilesN;
    const int blocks = (waves + 7) / 8;      // 8 waves (256 threads) / block
    gemm_wmma_f32<<<blocks, 256, 0, s>>>(A, Bw, bias, C, M, N, K, tilesM, tilesN);
}

extern "C" void kernel_launch(void* const* d_in, const int* in_sizes, int n_in,
                              void* d_out, int out_size, void* d_ws, size_t ws_size,
                              hipStream_t stream)
{
    const float* query  = (const float*)d_in[0];
    const float* refp   = (const float*)d_in[1];
    const float* value  = (const float*)d_in[2];
    const float* W_off  = (const float*)d_in[3];
    const float* b_off  = (const float*)d_in[4];
    const float* W_attn = (const float*)d_in[5];
    const float* b_attn = (const float*)d_in[6];
    const float* W_val  = (const float*)d_in[7];
    const float* b_val  = (const float*)d_in[8];
    const float* W_out  = (const float*)d_in[9];
    const float* b_out  = (const float*)d_in[10];
    float* out = (float*)d_out;

    const int M = B_C * NQ_C;                 // 23062 rows

    float* vbuf    = (float*)d_ws;            // [M,256]
    float* offbuf  = vbuf   + (size_t)M * 256;// [M,256]
    float* attnbuf = offbuf + (size_t)M * 256;// [M,128]
    float* midbuf  = attnbuf + (size_t)M * 128;// [M,256]

    // 1) value projection
    launch_gemm(value, W_val, b_val, vbuf, M, 256, 256, stream);
    // 2) sampling-offset projection
    launch_gemm(query, W_off, b_off, offbuf, M, 256, 256, stream);
    // 3) attention-logit projection
    launch_gemm(query, W_attn, b_attn, attnbuf, M, 128, 256, stream);

    // 4) softmax + bilinear deformable sampling
    {
        const int waves  = B_C * NQ_C * NHEAD;
        const int blocks = (waves + 7) / 8;
        msda_sample_kernel<<<blocks, 256, 0, stream>>>(vbuf, offbuf, attnbuf,
                                                       refp, midbuf);
    }

    // 5) output projection
    launch_gemm(midbuf, W_out, b_out, out, M, 256, 256, stream);
}